// Axial_TransformerBlock_91328184582269
// MI455X (gfx1250) — compile-verified
//
#include <hip/hip_runtime.h>
#include <math.h>

// ---------------- CDNA5 WMMA f32 16x16x4 ----------------
typedef __attribute__((ext_vector_type(2))) float v2f;
typedef __attribute__((ext_vector_type(8))) float v8f;

__device__ __forceinline__ v8f wmma_f32x4(v2f a, v2f b, v8f c) {
  // (neg_a, A, neg_b, B, c_mod, C, reuse_a, reuse_b)
  return __builtin_amdgcn_wmma_f32_16x16x4_f32(false, a, false, b, (short)0, c, false, false);
}

// Sizes
#define DIMC 96
#define KSZ  96
#define HW   9216     // 96*96
#define NPIX 36864    // 4*9216
#define NCOL 384      // 4*96
#define HID  255
#define HID2 510
#define EPSF 1e-5f

// ---------------- LayerNorm over C (per pixel) ----------------
__global__ void ln_kernel(const float* __restrict__ x, const float* __restrict__ w,
                          const float* __restrict__ b, float* __restrict__ y) {
  int wave = threadIdx.x >> 5, lane = threadIdx.x & 31;
  int p = blockIdx.x * 8 + wave;                 // pixel id < 36864
  int n = p / HW, hw = p % HW;
  size_t base = (size_t)n * DIMC * HW + hw;
  float v0 = x[base + (size_t)lane * HW];
  float v1 = x[base + (size_t)(lane + 32) * HW];
  float v2 = x[base + (size_t)(lane + 64) * HW];
  float s = v0 + v1 + v2, sq = v0 * v0 + v1 * v1 + v2 * v2;
  #pragma unroll
  for (int m = 16; m >= 1; m >>= 1) {
    s  += __shfl_xor(s,  m, 32);
    sq += __shfl_xor(sq, m, 32);
  }
  float mean = s * (1.f / 96.f);
  float var  = sq * (1.f / 96.f) - mean * mean;
  float is   = rsqrtf(var + EPSF);
  y[base + (size_t)lane * HW]        = (v0 - mean) * is * w[lane]      + b[lane];
  y[base + (size_t)(lane + 32) * HW] = (v1 - mean) * is * w[lane + 32] + b[lane + 32];
  y[base + (size_t)(lane + 64) * HW] = (v2 - mean) * is * w[lane + 64] + b[lane + 64];
}

// ---------------- finalize BN stats ----------------
__global__ void finalize_stats(const float* __restrict__ s, const float* __restrict__ sq,
                               float* __restrict__ mean, float* __restrict__ istd,
                               int n, float cnt) {
  int i = blockIdx.x * blockDim.x + threadIdx.x;
  if (i < n) {
    float m = s[i] / cnt;
    float v = sq[i] / cnt - m * m;
    mean[i] = m;
    istd[i] = rsqrtf(v + EPSF);
  }
}

// ---------------- qkv GEMM: (192x96)@(96x96) per column, WMMA, 2-way N-block ----------------
// axis 0: column = (n, w), l = h.  axis 1: column = (n, h), l = w.
__global__ void qkv_gemm_kernel(const float* __restrict__ yin, const float* __restrict__ wqkv,
                                float* __restrict__ qkv, float* __restrict__ ssum,
                                float* __restrict__ ssq, int axis) {
  int wave = threadIdx.x >> 5, lane = threadIdx.x & 31;
  int t = blockIdx.x * 8 + wave;                 // < 13824 = 384 cols * 36 wave-tiles
  int col = t / 36, tt = t % 36, tm = tt / 3, tn2 = tt % 3;
  int lm = lane & 15, half = lane >> 4, kb = half * 2;
  int m  = tm * 16 + lm;                         // output channel (A row)
  int l0 = tn2 * 32 + lm;                        // L positions of the two B tiles
  int l1 = l0 + 16;
  int n = col / KSZ, sp = col % KSZ;
  size_t yb0, yb1;
  if (axis == 0) {
    yb0 = (size_t)n * DIMC * HW + (size_t)l0 * KSZ + sp;
    yb1 = (size_t)n * DIMC * HW + (size_t)l1 * KSZ + sp;
  } else {
    yb0 = (size_t)n * DIMC * HW + (size_t)sp * KSZ + l0;
    yb1 = (size_t)n * DIMC * HW + (size_t)sp * KSZ + l1;
  }
  v8f acc0 = {}, acc1 = {};
  for (int k0 = 0; k0 < 96; k0 += 4) {
    v2f a, b0, b1;
    a[0]  = wqkv[m * 96 + k0 + kb];
    a[1]  = wqkv[m * 96 + k0 + kb + 1];
    b0[0] = yin[yb0 + (size_t)(k0 + kb) * HW];
    b0[1] = yin[yb0 + (size_t)(k0 + kb + 1) * HW];
    b1[0] = yin[yb1 + (size_t)(k0 + kb) * HW];
    b1[1] = yin[yb1 + (size_t)(k0 + kb + 1) * HW];
    acc0 = wmma_f32x4(a, b0, acc0);
    acc1 = wmma_f32x4(a, b1, acc1);
  }
  size_t obase = (size_t)col * 192 * 96;
  #pragma unroll
  for (int r = 0; r < 8; ++r) {
    int row = tm * 16 + half * 8 + r;
    float v0 = acc0[r], v1 = acc1[r];
    qkv[obase + (size_t)row * 96 + (tn2 * 32 + lm)]      = v0;
    qkv[obase + (size_t)row * 96 + (tn2 * 32 + 16 + lm)] = v1;
    float sv = v0 + v1, sv2 = v0 * v0 + v1 * v1;
    #pragma unroll
    for (int msk = 8; msk >= 1; msk >>= 1) {
      sv  += __shfl_xor(sv,  msk, 32);
      sv2 += __shfl_xor(sv2, msk, 32);
    }
    if (lm == 0) { atomicAdd(&ssum[row], sv); atomicAdd(&ssq[row], sv2); }
  }
}

// ---------------- sim statistics pass (qk / qr / kr recomputed, stats only) ----------------
__global__ void sim_stats_kernel(const float* __restrict__ qkv, const float* __restrict__ rel,
                                 const float* __restrict__ mean, const float* __restrict__ istd,
                                 const float* __restrict__ gam, const float* __restrict__ bet,
                                 float* __restrict__ ssum, float* __restrict__ ssq) {
  __shared__ float qs[576], ks_[576], qe[1146], ke[1146], red[6];
  int col = blockIdx.x, g = blockIdx.y, tid = threadIdx.x;
  size_t qbase = (size_t)col * 192 * 96 + (size_t)g * 24 * 96;
  for (int e = tid; e < 576; e += blockDim.x) {
    int c = e / 96, l = e % 96;
    int oq = g * 24 + c, ok = g * 24 + 6 + c;
    qs[e]  = (qkv[qbase + c * 96 + l]       - mean[oq]) * istd[oq] * gam[oq] + bet[oq];
    ks_[e] = (qkv[qbase + (6 + c) * 96 + l] - mean[ok]) * istd[ok] * gam[ok] + bet[ok];
  }
  for (int e = tid; e < 1146; e += blockDim.x) { qe[e] = rel[e]; ke[e] = rel[1146 + e]; }
  if (tid < 6) red[tid] = 0.f;
  __syncthreads();
  float aqk = 0, aqk2 = 0, aqr = 0, aqr2 = 0, akr = 0, akr2 = 0;
  for (int e = tid; e < 9216; e += blockDim.x) {
    int i = e / 96, j = e % 96, d = i - j + 95;
    float qk = 0, qr = 0, kr = 0;
    #pragma unroll
    for (int c = 0; c < 6; ++c) {
      float qi = qs[c * 96 + i];
      qk += qi * ks_[c * 96 + j];
      qr += qi * qe[c * 191 + d];
      kr += ks_[c * 96 + i] * ke[c * 191 + d];   // kr_pre[i][j]
    }
    aqk += qk; aqk2 += qk * qk; aqr += qr; aqr2 += qr * qr; akr += kr; akr2 += kr * kr;
  }
  atomicAdd(&red[0], aqk); atomicAdd(&red[1], aqk2);
  atomicAdd(&red[2], aqr); atomicAdd(&red[3], aqr2);
  atomicAdd(&red[4], akr); atomicAdd(&red[5], akr2);
  __syncthreads();
  if (tid == 0) {
    atomicAdd(&ssum[g],      red[0]); atomicAdd(&ssq[g],      red[1]);
    atomicAdd(&ssum[8 + g],  red[2]); atomicAdd(&ssq[8 + g],  red[3]);
    atomicAdd(&ssum[16 + g], red[4]); atomicAdd(&ssq[16 + g], red[5]);
  }
}

// ---------------- attention: normalized sim, online softmax, sv/sve + out stats ----------------
__global__ void attn_kernel(const float* __restrict__ qkv, const float* __restrict__ rel,
                            const float* __restrict__ qmean, const float* __restrict__ qistd,
                            const float* __restrict__ qg, const float* __restrict__ qb,
                            const float* __restrict__ smean, const float* __restrict__ sistd,
                            const float* __restrict__ sg, const float* __restrict__ sb,
                            float* __restrict__ sv, float* __restrict__ sve,
                            float* __restrict__ osum, float* __restrict__ osq) {
  __shared__ float qs[576], ks_[576], vs[1152], qe[1146], ke[1146], ve[2292], oacc[48];
  int col = blockIdx.x, g = blockIdx.y, tid = threadIdx.x;  // blockDim = 96
  size_t qbase = (size_t)col * 192 * 96 + (size_t)g * 24 * 96;
  for (int e = tid; e < 576; e += 96) {
    int c = e / 96, l = e % 96;
    int oq = g * 24 + c, ok = g * 24 + 6 + c;
    qs[e]  = (qkv[qbase + c * 96 + l]       - qmean[oq]) * qistd[oq] * qg[oq] + qb[oq];
    ks_[e] = (qkv[qbase + (6 + c) * 96 + l] - qmean[ok]) * qistd[ok] * qg[ok] + qb[ok];
  }
  for (int e = tid; e < 1152; e += 96) {
    int c = e / 96, ov = g * 24 + 12 + c;
    vs[e] = (qkv[qbase + (12 + c) * 96 + (e % 96)] - qmean[ov]) * qistd[ov] * qg[ov] + qb[ov];
  }
  for (int e = tid; e < 1146; e += 96) { qe[e] = rel[e]; ke[e] = rel[1146 + e]; }
  for (int e = tid; e < 2292; e += 96) ve[e] = rel[2292 + e];
  if (tid < 48) oacc[tid] = 0.f;
  __syncthreads();

  int i = tid;
  float m0 = smean[g],      is0 = sistd[g]      * sg[g],      b0 = sb[g];
  float m1 = smean[8 + g],  is1 = sistd[8 + g]  * sg[8 + g],  b1 = sb[8 + g];
  float m2 = smean[16 + g], is2 = sistd[16 + g] * sg[16 + g], b2 = sb[16 + g];

  auto sim_ij = [&](int ii, int jj) -> float {
    int d = ii - jj + 95, d2 = jj - ii + 95;
    float qk = 0, qr = 0, kr = 0;
    #pragma unroll
    for (int c = 0; c < 6; ++c) {
      float qi = qs[c * 96 + ii], kj = ks_[c * 96 + jj];
      qk += qi * kj;
      qr += qi * qe[c * 191 + d];
      kr += kj * ke[c * 191 + d2];               // kr^T[i][j]
    }
    return (qk - m0) * is0 + b0 + (qr - m1) * is1 + b1 + (kr - m2) * is2 + b2;
  };

  float mx = -1e30f, den = 0.f;
  for (int j = 0; j < 96; ++j) {
    float s = sim_ij(i, j);
    float nm = fmaxf(mx, s);
    den = den * __expf(mx - nm) + __expf(s - nm);
    mx = nm;
  }
  float inv = 1.f / den;
  float svA[12] = {}, sveA[12] = {};
  for (int j = 0; j < 96; ++j) {
    float p = __expf(sim_ij(i, j) - mx) * inv;
    int d = i - j + 95;
    #pragma unroll
    for (int c = 0; c < 12; ++c) {
      svA[c]  += p * vs[c * 96 + j];
      sveA[c] += p * ve[c * 191 + d];
    }
  }
  size_t obase = ((size_t)(col * 8 + g) * 12) * 96;
  #pragma unroll
  for (int c = 0; c < 12; ++c) {
    sv[obase + c * 96 + i]  = svA[c];
    sve[obase + c * 96 + i] = sveA[c];
    atomicAdd(&oacc[(2 * c) * 2],     svA[c]);
    atomicAdd(&oacc[(2 * c) * 2 + 1], svA[c] * svA[c]);
    atomicAdd(&oacc[(2 * c + 1) * 2],     sveA[c]);
    atomicAdd(&oacc[(2 * c + 1) * 2 + 1], sveA[c] * sveA[c]);
  }
  __syncthreads();
  if (tid < 24) {
    int och = g * 24 + tid;
    atomicAdd(&osum[och], oacc[tid * 2]);
    atomicAdd(&osq[och],  oacc[tid * 2 + 1]);
  }
}

// ---------------- combine: BN(sv)+BN(sve), write back to NCHW ----------------
__global__ void combine_kernel(const float* __restrict__ sv, const float* __restrict__ sve,
                               const float* __restrict__ om, const float* __restrict__ ois,
                               const float* __restrict__ og, const float* __restrict__ ob,
                               float* __restrict__ outax, int axis) {
  int idx = blockIdx.x * 256 + threadIdx.x;      // < 3538944
  int i = idx % 96;
  int c = (idx / 96) % 12;
  int g = (idx / (96 * 12)) % 8;
  int col = idx / (96 * 12 * 8);
  size_t sbase = ((size_t)(col * 8 + g) * 12 + c) * 96 + i;
  int o0 = g * 24 + 2 * c, o1 = o0 + 1;
  float a = (sv[sbase]  - om[o0]) * ois[o0] * og[o0] + ob[o0];
  float b = (sve[sbase] - om[o1]) * ois[o1] * og[o1] + ob[o1];
  int n = col / KSZ, sp = col % KSZ, cout = g * 12 + c;
  size_t oi = (size_t)n * DIMC * HW + (size_t)cout * HW +
              (axis == 0 ? (size_t)i * KSZ + sp : (size_t)sp * KSZ + i);
  outax[oi] = a + b;
}

// ---------------- elementwise add ----------------
__global__ void add_kernel(const float* __restrict__ a, const float* __restrict__ b,
                           float* __restrict__ o) {
  int idx = blockIdx.x * 256 + threadIdx.x;
  o[idx] = a[idx] + b[idx];
}

// ---------------- FFN-in GEMM: (510x96)@(96x36864), WMMA, 4-way N-block ----------------
__global__ void ffn1_gemm_kernel(const float* __restrict__ y2, const float* __restrict__ win,
                                 float* __restrict__ h) {
  int wave = threadIdx.x >> 5, lane = threadIdx.x & 31;
  int t = blockIdx.x * 8 + wave;                 // < 18432 = 32 * 576
  int tm = t / 576, tn4 = t % 576;
  int lm = lane & 15, half = lane >> 4, kb = half * 2;
  int m  = tm * 16 + lm;
  int p0 = tn4 * 64;                             // 64-pixel block (never crosses batch: 9216%64==0)
  int n = p0 / HW, hw0 = p0 % HW;
  size_t bbase = (size_t)n * DIMC * HW + hw0 + lm;
  v8f acc0 = {}, acc1 = {}, acc2 = {}, acc3 = {};
  for (int k0 = 0; k0 < 96; k0 += 4) {
    size_t kadr = bbase + (size_t)(k0 + kb) * HW;
    if (k0 + 16 < 96)                            // prefetch next K-panel of B (36 KB stride)
      __builtin_prefetch(&y2[bbase + (size_t)(k0 + 16 + kb) * HW], 0, 1);
    v2f a, b0, b1, b2, b3;
    a[0]  = (m < HID2) ? win[m * 96 + k0 + kb] : 0.f;
    a[1]  = (m < HID2) ? win[m * 96 + k0 + kb + 1] : 0.f;
    b0[0] = y2[kadr];           b0[1] = y2[kadr + HW];
    b1[0] = y2[kadr + 16];      b1[1] = y2[kadr + HW + 16];
    b2[0] = y2[kadr + 32];      b2[1] = y2[kadr + HW + 32];
    b3[0] = y2[kadr + 48];      b3[1] = y2[kadr + HW + 48];
    acc0 = wmma_f32x4(a, b0, acc0);
    acc1 = wmma_f32x4(a, b1, acc1);
    acc2 = wmma_f32x4(a, b2, acc2);
    acc3 = wmma_f32x4(a, b3, acc3);
  }
  #pragma unroll
  for (int r = 0; r < 8; ++r) {
    int row = tm * 16 + half * 8 + r;
    if (row < HID2) {
      size_t ob = (size_t)row * NPIX + p0 + lm;
      h[ob]      = acc0[r];
      h[ob + 16] = acc1[r];
      h[ob + 32] = acc2[r];
      h[ob + 48] = acc3[r];
    }
  }
}

// ---------------- depthwise 3x3 SAME + GEGLU gate ----------------
__global__ void dwgate_kernel(const float* __restrict__ h, const float* __restrict__ wdw,
                              float* __restrict__ g) {
  int idx = blockIdx.x * 256 + threadIdx.x;      // < 255*36864 = 9400320
  int p = idx % NPIX, o2 = idx / NPIX;
  int n = p / HW, hw = p % HW, yy = hw / 96, xx = hw % 96;
  float a = 0.f, b = 0.f;
  const float* wa = wdw + o2 * 9;
  const float* wb = wdw + (o2 + HID) * 9;
  size_t basea = (size_t)o2 * NPIX + (size_t)n * HW;
  size_t baseb = (size_t)(o2 + HID) * NPIX + (size_t)n * HW;
  #pragma unroll
  for (int dy = -1; dy <= 1; ++dy) {
    int yyy = yy + dy;
    if (yyy < 0 || yyy >= 96) continue;
    #pragma unroll
    for (int dx = -1; dx <= 1; ++dx) {
      int xxx = xx + dx;
      if (xxx < 0 || xxx >= 96) continue;
      int off = yyy * 96 + xxx;
      a += h[basea + off] * wa[(dy + 1) * 3 + dx + 1];
      b += h[baseb + off] * wb[(dy + 1) * 3 + dx + 1];
    }
  }
  float ge = 0.5f * a * (1.f + erff(a * 0.70710678118f));   // exact GELU
  g[idx] = ge * b;
}

// ---------------- FFN-out GEMM: (96x255)@(255x36864) + residual, WMMA, 4-way N-block ----------------
__global__ void ffn2_gemm_kernel(const float* __restrict__ g, const float* __restrict__ wout,
                                 const float* __restrict__ x1, float* __restrict__ out) {
  int wave = threadIdx.x >> 5, lane = threadIdx.x & 31;
  int t = blockIdx.x * 8 + wave;                 // < 3456 = 6 * 576
  int tm = t / 576, tn4 = t % 576;
  int lm = lane & 15, half = lane >> 4, kb = half * 2;
  int m  = tm * 16 + lm;                         // < 96
  int p0 = tn4 * 64;
  v8f acc0 = {}, acc1 = {}, acc2 = {}, acc3 = {};
  for (int k0 = 0; k0 < 256; k0 += 4) {
    int k1 = k0 + kb, k2 = k0 + kb + 1;
    bool v1 = (k1 < HID), v2 = (k2 < HID);
    size_t g1 = (size_t)k1 * NPIX + p0 + lm;
    size_t g2 = (size_t)k2 * NPIX + p0 + lm;
    v2f a, b0, b1, b2, b3;
    a[0]  = v1 ? wout[m * HID + k1] : 0.f;
    a[1]  = v2 ? wout[m * HID + k2] : 0.f;
    b0[0] = v1 ? g[g1]      : 0.f;  b0[1] = v2 ? g[g2]      : 0.f;
    b1[0] = v1 ? g[g1 + 16] : 0.f;  b1[1] = v2 ? g[g2 + 16] : 0.f;
    b2[0] = v1 ? g[g1 + 32] : 0.f;  b2[1] = v2 ? g[g2 + 32] : 0.f;
    b3[0] = v1 ? g[g1 + 48] : 0.f;  b3[1] = v2 ? g[g2 + 48] : 0.f;
    acc0 = wmma_f32x4(a, b0, acc0);
    acc1 = wmma_f32x4(a, b1, acc1);
    acc2 = wmma_f32x4(a, b2, acc2);
    acc3 = wmma_f32x4(a, b3, acc3);
  }
  int n = p0 / HW, hw0 = p0 % HW;
  #pragma unroll
  for (int r = 0; r < 8; ++r) {
    int row = tm * 16 + half * 8 + r;
    size_t oi = (size_t)n * DIMC * HW + (size_t)row * HW + hw0 + lm;
    out[oi]      = x1[oi]      + acc0[r];
    out[oi + 16] = x1[oi + 16] + acc1[r];
    out[oi + 32] = x1[oi + 32] + acc2[r];
    out[oi + 48] = x1[oi + 48] + acc3[r];
  }
}

// ---------------- host orchestration ----------------
extern "C" void kernel_launch(void* const* d_in, const int* in_sizes, int n_in,
                              void* d_out, int out_size, void* d_ws, size_t ws_size,
                              hipStream_t stream) {
  const float* x     = (const float*)d_in[0];
  const float* ln1w  = (const float*)d_in[1];
  const float* ln1b  = (const float*)d_in[2];
  const float* ln2w  = (const float*)d_in[19];
  const float* ln2b  = (const float*)d_in[20];
  const float* win   = (const float*)d_in[21];
  const float* wdw   = (const float*)d_in[22];
  const float* wout  = (const float*)d_in[23];

  float* W = (float*)d_ws;
  // Arena: A reused for axial temporaries then FFN hidden; B ln-out/axial-out/y2; C x1; D gated
  float* bufA   = W;                         // 18,800,640 floats
  float* bufB   = W + 18800640;              //  3,538,944
  float* bufC   = bufB + 3538944;            //  3,538,944
  float* bufD   = bufC + 3538944;            //  9,400,320
  float* S      = bufD + 9400320;            //  2,048 stats
  float* qkvBuf = bufA;                      //  7,077,888
  float* svBuf  = bufA + 7077888;            //  3,538,944
  float* sveBuf = bufA + 7077888 + 3538944;  //  3,538,944
  float* axBuf  = bufA + 7077888 + 2 * 3538944;

  float *qS = S,       *qQ = S + 192,  *qM = S + 384,  *qI = S + 576;
  float *sS = S + 768, *sQ = S + 792,  *sM = S + 816,  *sI = S + 840;
  float *oS = S + 864, *oQ = S + 1056, *oM = S + 1248, *oI = S + 1440;

  auto axial = [&](const float* in, float* outax, int base, int axis) {
    const float* wq  = (const float*)d_in[base + 0];
    const float* gq  = (const float*)d_in[base + 1];
    const float* bq  = (const float*)d_in[base + 2];
    const float* rel = (const float*)d_in[base + 3];
    const float* gs  = (const float*)d_in[base + 4];
    const float* bs  = (const float*)d_in[base + 5];
    const float* go  = (const float*)d_in[base + 6];
    const float* bo  = (const float*)d_in[base + 7];
    hipMemsetAsync(S, 0, 2048 * sizeof(float), stream);
    qkv_gemm_kernel<<<1728, 256, 0, stream>>>(in, wq, qkvBuf, qS, qQ, axis);
    finalize_stats<<<1, 192, 0, stream>>>(qS, qQ, qM, qI, 192, 36864.f);
    sim_stats_kernel<<<dim3(384, 8), 256, 0, stream>>>(qkvBuf, rel, qM, qI, gq, bq, sS, sQ);
    finalize_stats<<<1, 32, 0, stream>>>(sS, sQ, sM, sI, 24, 3538944.f);
    attn_kernel<<<dim3(384, 8), 96, 0, stream>>>(qkvBuf, rel, qM, qI, gq, bq,
                                                 sM, sI, gs, bs, svBuf, sveBuf, oS, oQ);
    finalize_stats<<<1, 192, 0, stream>>>(oS, oQ, oM, oI, 192, 36864.f);
    combine_kernel<<<13824, 256, 0, stream>>>(svBuf, sveBuf, oM, oI, go, bo, outax, axis);
  };

  // y = LN1(x)
  ln_kernel<<<4608, 256, 0, stream>>>(x, ln1w, ln1b, bufB);
  // y = axial_h(y)  (in-place on B: input consumed by qkv GEMM before combine writes)
  axial(bufB, bufB, 3, /*axis=*/0);
  // y = axial_w(y)
  axial(bufB, axBuf, 11, /*axis=*/1);
  // x1 = x + y
  add_kernel<<<13824, 256, 0, stream>>>(x, axBuf, bufC);
  // y2 = LN2(x1)
  ln_kernel<<<4608, 256, 0, stream>>>(bufC, ln2w, ln2b, bufB);
  // h = win @ y2   (overwrites dead axial buffers in A)
  ffn1_gemm_kernel<<<2304, 256, 0, stream>>>(bufB, win, bufA);
  // g = gelu(dw(h)[:255]) * dw(h)[255:]
  dwgate_kernel<<<36720, 256, 0, stream>>>(bufA, wdw, bufD);
  // out = x1 + wout @ g
  ffn2_gemm_kernel<<<432, 256, 0, stream>>>(bufD, wout, bufC, (float*)d_out);
}